// SSIMLoss_25477746000488
// MI455X (gfx1250) — compile-verified
//
#include <hip/hip_runtime.h>

// ---------------------------------------------------------------------------
// Fused 3D-SSIM (win=7) for (4,1,128,128,128) f32 volumes on gfx1250.
//
// The 7-tap box filter along an axis is a banded matmul executed on the
// tensor path:  V_WMMA_F32_16X16X4_F32,  Out(16x16) = In(16x24) x Band(24x16),
// Band[k][n] = 1/7 iff n <= k <= n+6 (zero for pad rows k=22,23).
// One fused kernel keeps all intermediates in LDS; HBM traffic is the 67 MB
// compulsory input read (~2.9 us floor at 23.3 TB/s); tile halo re-reads are
// absorbed by the 192 MB L2.
// ---------------------------------------------------------------------------

typedef float v2f __attribute__((ext_vector_type(2)));
typedef float v8f __attribute__((ext_vector_type(8)));

#define TD   2            // output depth per tile (61 tiles cover 122 exactly)
#define DZ   (TD + 6)     // 8 input d-slices
#define HB   22           // input block extent in h
#define WB   22           // input block extent in w
#define NROW (DZ * HB)    // 176 flattened (dz,h) rows -> 11 groups of 16
#define NGRP (NROW / 16)
#define COVN (343.0f / 342.0f)
#define OUTD 122

__device__ __forceinline__ float bandv(int k, int n) {
    // 7-tap VALID box band matrix (already includes the 1/7 of one pass).
    return (k >= n && k <= n + 6) ? (1.0f / 7.0f) : 0.0f;
}

// One quantity's full separable pipeline, specialized at compile time so the
// product selection is straight-line code (no runtime switch in the K loop).
template <int Q>
__device__ __forceinline__ void process_q(
    const float* sx, const float* sy, float* wbuf, float* hbuf, float* out5,
    int wave, int lane)
{
    const int mrow = lane & 15;              // M index (A) == N index (B/D)
    const int khi  = (lane < 16) ? 0 : 2;    // K sub-offset per A/B layout
    const int half = (lane < 16) ? 0 : 8;    // D-fragment M offset

    // ---- W-pass: banded matmul along w (V_WMMA_F32_16X16X4_F32) ----------
    for (int g = wave; g < NGRP; g += 4) {             // scalar-uniform loop
        const int row  = g * 16 + mrow;                // (dz,h) row, < 176
        const int base = row * WB;
        v8f c = {};
        #pragma unroll
        for (int kc = 0; kc < 6; ++kc) {
            const int k0  = kc * 4 + khi;
            const int wA0 = min(k0, WB - 1);           // pad cols: B == 0
            const int wA1 = min(k0 + 1, WB - 1);
            v2f a, bb;
            if (Q == 0) {
                a.x = sx[base + wA0];              a.y = sx[base + wA1];
            } else if (Q == 1) {
                a.x = sy[base + wA0];              a.y = sy[base + wA1];
            } else if (Q == 2) {
                float x0 = sx[base + wA0], x1 = sx[base + wA1];
                a.x = x0 * x0;                     a.y = x1 * x1;
            } else if (Q == 3) {
                float y0 = sy[base + wA0], y1 = sy[base + wA1];
                a.x = y0 * y0;                     a.y = y1 * y1;
            } else {
                float x0 = sx[base + wA0], x1 = sx[base + wA1];
                float y0 = sy[base + wA0], y1 = sy[base + wA1];
                a.x = x0 * y0;                     a.y = x1 * y1;
            }
            bb.x = bandv(k0,     mrow);
            bb.y = bandv(k0 + 1, mrow);
            c = __builtin_amdgcn_wmma_f32_16x16x4_f32(
                    false, a, false, bb, (short)0, c, false, false);
        }
        // D layout: VGPR r -> M = r (lanes 0-15) / r+8 (lanes 16-31)
        const int rbase = g * 16 + half;
        #pragma unroll
        for (int r = 0; r < 8; ++r)
            wbuf[(rbase + r) * 16 + mrow] = c[r];
    }
    __syncthreads();

    // ---- H-pass: banded matmul along h ------------------------------------
    for (int dz = wave; dz < DZ; dz += 4) {            // scalar-uniform loop
        v8f c = {};
        #pragma unroll
        for (int kc = 0; kc < 6; ++kc) {
            const int k0  = kc * 4 + khi;
            const int h0a = min(k0, HB - 1);
            const int h1a = min(k0 + 1, HB - 1);
            v2f a, bb;
            a.x  = wbuf[(dz * HB + h0a) * 16 + mrow];  // A[m=w][k=h]
            a.y  = wbuf[(dz * HB + h1a) * 16 + mrow];
            bb.x = bandv(k0,     mrow);
            bb.y = bandv(k0 + 1, mrow);
            c = __builtin_amdgcn_wmma_f32_16x16x4_f32(
                    false, a, false, bb, (short)0, c, false, false);
        }
        #pragma unroll
        for (int r = 0; r < 8; ++r)
            hbuf[dz * 256 + (half + r) * 16 + mrow] = c[r];   // [w*16 + h]
    }
    __syncthreads();

    // ---- D-pass: 7-tap sum along d (cheap VALU) ---------------------------
    const int tid = wave * 32 + lane;
    for (int e = tid; e < TD * 256; e += 128) {
        const int dd = e >> 8, rem = e & 255;
        float s = 0.0f;
        #pragma unroll
        for (int j = 0; j < 7; ++j) s += hbuf[(dd + j) * 256 + rem];
        out5[Q * (TD * 256) + e] = s * (1.0f / 7.0f);
    }
    __syncthreads();
}

__global__ __launch_bounds__(128)
void ssim3d_tile_kernel(const float* __restrict__ X, const float* __restrict__ Y,
                        const float* __restrict__ DR, float* __restrict__ partial)
{
    __shared__ float sx[DZ * HB * WB];      // 3872 f32
    __shared__ float sy[DZ * HB * WB];      // 3872 f32
    __shared__ float wbuf[NROW * 16];       // 2816 f32  (w-filtered, one q)
    __shared__ float hbuf[DZ * 256];        // 2048 f32  (w+h filtered, one q)
    __shared__ float out5[5 * TD * 256];    // 2560 f32  (all 5 filtered maps)
    __shared__ float red[128];

    const int tid  = threadIdx.x;
    // readfirstlane: provably wave-uniform -> scalar loops, no EXEC masking
    // around the WMMA region (WMMA requires EXEC all-ones).
    const int wave = __builtin_amdgcn_readfirstlane(tid >> 5);
    const int lane = tid & 31;

    const int wt = blockIdx.x & 7;
    const int ht = blockIdx.x >> 3;
    const int b  = blockIdx.z;
    const int w0 = wt * 16, h0 = ht * 16, d0 = blockIdx.y * TD;

    // ---- Phase A: cooperative load of the halo block ----------------------
    // Edge tiles clamp h/w to 127; clamped voxels only feed masked outputs.
    for (int e = tid; e < DZ * HB * WB; e += 128) {
        int iz = e / (HB * WB);
        int r  = e % (HB * WB);
        int ih = r / WB, iw = r % WB;
        int gz = d0 + iz;                    // max 120+7 = 127, always valid
        int gh = min(h0 + ih, 127);
        int gw = min(w0 + iw, 127);
        size_t gi = (((size_t)b * 128 + gz) * 128 + gh) * 128 + gw;
        sx[e] = X[gi];
        sy[e] = Y[gi];
    }
    __syncthreads();

    process_q<0>(sx, sy, wbuf, hbuf, out5, wave, lane);   // ux
    process_q<1>(sx, sy, wbuf, hbuf, out5, wave, lane);   // uy
    process_q<2>(sx, sy, wbuf, hbuf, out5, wave, lane);   // uxx
    process_q<3>(sx, sy, wbuf, hbuf, out5, wave, lane);   // uyy
    process_q<4>(sx, sy, wbuf, hbuf, out5, wave, lane);   // uxy

    // ---- Phase E: SSIM + deterministic block reduction --------------------
    const float dr = DR[b];
    const float C1 = (0.01f * dr) * (0.01f * dr);
    const float C2 = (0.03f * dr) * (0.03f * dr);

    float acc = 0.0f;
    for (int e = tid; e < TD * 256; e += 128) {
        int rem = e & 255;
        int w = rem >> 4, h = rem & 15;
        if ((w0 + w) < OUTD && (h0 + h) < OUTD) {        // d always valid
            float ux  = out5[0 * (TD * 256) + e];
            float uy  = out5[1 * (TD * 256) + e];
            float uxx = out5[2 * (TD * 256) + e];
            float uyy = out5[3 * (TD * 256) + e];
            float uxy = out5[4 * (TD * 256) + e];
            float vx  = COVN * (uxx - ux * ux);
            float vy  = COVN * (uyy - uy * uy);
            float vxy = COVN * (uxy - ux * uy);
            float A1 = 2.0f * ux * uy + C1;
            float A2 = 2.0f * vxy + C2;
            float B1 = ux * ux + uy * uy + C1;
            float B2 = vx + vy + C2;
            acc += (A1 * A2) / (B1 * B2);
        }
    }
    red[tid] = acc;
    __syncthreads();
    #pragma unroll
    for (int s = 64; s > 0; s >>= 1) {
        if (tid < s) red[tid] += red[tid + s];
        __syncthreads();
    }
    if (tid == 0)
        partial[((size_t)blockIdx.z * 61 + blockIdx.y) * 64 + blockIdx.x] = red[0];
}

__global__ __launch_bounds__(256)
void ssim3d_reduce_kernel(const float* __restrict__ partial, int n,
                          float* __restrict__ out)
{
    __shared__ float red[256];
    float a = 0.0f;
    for (int i = threadIdx.x; i < n; i += 256) a += partial[i];   // fixed order
    red[threadIdx.x] = a;
    __syncthreads();
    #pragma unroll
    for (int s = 128; s > 0; s >>= 1) {
        if (threadIdx.x < s) red[threadIdx.x] += red[threadIdx.x + s];
        __syncthreads();
    }
    if (threadIdx.x == 0)
        out[0] = red[0] * (1.0f / (4.0f * 122.0f * 122.0f * 122.0f));
}

extern "C" void kernel_launch(void* const* d_in, const int* in_sizes, int n_in,
                              void* d_out, int out_size, void* d_ws, size_t ws_size,
                              hipStream_t stream) {
    const float* X  = (const float*)d_in[0];
    const float* Y  = (const float*)d_in[1];
    const float* DR = (const float*)d_in[2];      // (4,1,1,1,1) data_range
    float* partial  = (float*)d_ws;               // 15616 f32 block partials

    dim3 grid(64, 61, 4);                         // (w,h) tiles, d tiles, batch
    dim3 block(128);                              // 4 waves of 32
    ssim3d_tile_kernel<<<grid, block, 0, stream>>>(X, Y, DR, partial);
    ssim3d_reduce_kernel<<<1, 256, 0, stream>>>(partial, 64 * 61 * 4,
                                                (float*)d_out);
}